// RoadGNN_21964462752238
// MI455X (gfx1250) — compile-verified
//
#include <hip/hip_runtime.h>
#include <hip/hip_bf16.h>
#include <stdint.h>

#define NNODES 50000
#define NEDGES 800000
#define IN_DIM 64
#define HID 256
#define NEG_SLOPE 0.2f

typedef __attribute__((ext_vector_type(2))) float v2f;
typedef __attribute__((ext_vector_type(8))) float v8f;

// ---------------- order-preserving float<->uint for atomicMax segment-max ----
__device__ __forceinline__ unsigned enc_f32(float x) {
  unsigned b = __float_as_uint(x);
  return (b & 0x80000000u) ? ~b : (b | 0x80000000u);
}
__device__ __forceinline__ float dec_f32(unsigned k) {
  unsigned b = (k & 0x80000000u) ? (k ^ 0x80000000u) : ~k;
  return __uint_as_float(b);
}

// ---------------- H[M x 256] = X[M x K] @ W[K x 256] via V_WMMA_F32_16X16X4 --
// One wave computes one 16x16 output tile. M = 50000 = 3125*16 exactly,
// N = 256 = 16*16 exactly -> no partial tiles, EXEC stays all-ones.
__global__ __launch_bounds__(256) void gemm_wmma_f32(
    const float* __restrict__ X, const float* __restrict__ W,
    float* __restrict__ H, int K) {
  const int wave  = (blockIdx.x * blockDim.x + threadIdx.x) >> 5;
  const int lane  = threadIdx.x & 31;
  const int half  = lane >> 4;     // 0: lanes 0-15, 1: lanes 16-31
  const int l15   = lane & 15;
  const int tileN = wave & 15;     // 16 column tiles
  const int tileM = wave >> 4;     // 3125 row tiles
  const int row   = tileM * 16 + l15;   // A-matrix row for this lane
  const int col   = tileN * 16 + l15;   // B/C/D column for this lane

  v8f c = {};
  const float* xrow = X + (size_t)row * K;
  for (int k0 = 0; k0 < K; k0 += 4) {
    const int ka = k0 + 2 * half;     // lanes 0-15: K=k0,k0+1 ; 16-31: K=k0+2,k0+3
    v2f a, b;
    a.x = xrow[ka];
    a.y = xrow[ka + 1];
    b.x = W[(size_t)ka * HID + col];
    b.y = W[(size_t)(ka + 1) * HID + col];
    c = __builtin_amdgcn_wmma_f32_16x16x4_f32(
        /*neg_a=*/false, a, /*neg_b=*/false, b,
        /*c_mod=*/(short)0, c, /*reuse_a=*/false, /*reuse_b=*/false);
  }
  // C/D layout: VGPR r -> M = r (lanes 0-15) or M = 8 + r (lanes 16-31)
  const int mbase = tileM * 16 + half * 8;
  #pragma unroll
  for (int r = 0; r < 8; ++r)
    H[(size_t)(mbase + r) * HID + col] = c[r];
}

// ---------------- per-node dots: a_src[n] = h[n].att_src ; a_dst likewise ----
__global__ __launch_bounds__(256) void node_dots(
    const float* __restrict__ H,
    const float* __restrict__ att_src, const float* __restrict__ att_dst,
    float* __restrict__ a_src, float* __restrict__ a_dst) {
  const int node = (blockIdx.x * blockDim.x + threadIdx.x) >> 5;
  const int lane = threadIdx.x & 31;
  if (node >= NNODES) return;
  const float* hrow = H + (size_t)node * HID;
  float s = 0.f, d = 0.f;
  #pragma unroll
  for (int c = lane; c < HID; c += 32) {
    const float h = hrow[c];
    s += h * att_src[c];
    d += h * att_dst[c];
  }
  #pragma unroll
  for (int off = 16; off; off >>= 1) {
    s += __shfl_xor(s, off, 32);
    d += __shfl_xor(d, off, 32);
  }
  if (lane == 0) { a_src[node] = s; a_dst[node] = d; }
}

// ---------------- tiny: we_att = We @ att_e for both layers (6 floats each) --
__global__ void edge_weight_proj(const float* __restrict__ We1, const float* __restrict__ ae1,
                                 const float* __restrict__ We2, const float* __restrict__ ae2,
                                 float* __restrict__ we_att) {
  const int t = threadIdx.x;
  if (t < 6) {
    float s = 0.f;
    for (int k = 0; k < HID; ++k) s += We1[t * HID + k] * ae1[k];
    we_att[t] = s;
  } else if (t >= 8 && t < 14) {
    const int d = t - 8;
    float s = 0.f;
    for (int k = 0; k < HID; ++k) s += We2[d * HID + k] * ae2[k];
    we_att[8 + d] = s;
  }
}

// ---------------- zero-init helper --------------------------------------------
__global__ void zero_f32(float* __restrict__ p, int n) {
  const int i = blockIdx.x * blockDim.x + threadIdx.x;
  if (i < n) p[i] = 0.0f;
}

// ---------------- per-edge raw attention + segment max ------------------------
__global__ __launch_bounds__(256) void edge_alpha(
    const int* __restrict__ src, const int* __restrict__ dst,
    const float* __restrict__ edge_attr, const float* __restrict__ we_att,
    const float* __restrict__ a_src, const float* __restrict__ a_dst,
    float* __restrict__ alpha, unsigned* __restrict__ amax) {
  const int e = blockIdx.x * blockDim.x + threadIdx.x;
  if (e >= NEDGES) return;
  const float* ea = edge_attr + (size_t)e * 6;
  const float ae = ea[0]*we_att[0] + ea[1]*we_att[1] + ea[2]*we_att[2]
                 + ea[3]*we_att[3] + ea[4]*we_att[4] + ea[5]*we_att[5];
  const int d = dst[e];
  float a = a_src[src[e]] + a_dst[d] + ae;
  a = (a > 0.f) ? a : NEG_SLOPE * a;          // leaky_relu
  alpha[e] = a;
  atomicMax(amax + d, enc_f32(a));
}

// ---------------- exp(alpha - max) and denominator ----------------------------
__global__ __launch_bounds__(256) void edge_exp(
    const int* __restrict__ dst, float* __restrict__ alpha,
    const unsigned* __restrict__ amax, float* __restrict__ denom) {
  const int e = blockIdx.x * blockDim.x + threadIdx.x;
  if (e >= NEDGES) return;
  const int d = dst[e];
  const float ex = expf(alpha[e] - dec_f32(amax[d]));
  alpha[e] = ex;                               // reuse buffer as ex
  atomicAdd(denom + d, ex);
}

// ---------------- weighted scatter: acc[dst] += (ex/denom) * h[src] -----------
// one thread per (edge, 4-channel chunk): float4 load + 4 f32 atomic adds
__global__ __launch_bounds__(256) void edge_scatter(
    const int* __restrict__ src, const int* __restrict__ dst,
    const float* __restrict__ alpha, const float* __restrict__ denom,
    const float* __restrict__ H, float* __restrict__ acc) {
  const long long gid = (long long)blockIdx.x * blockDim.x + threadIdx.x;
  const int e = (int)(gid >> 6);
  if (e >= NEDGES) return;
  const int q = (int)(gid & 63);
  const int d = dst[e];
  const float w = alpha[e] / (denom[d] + 1e-16f);
  const float4 hv = ((const float4*)H)[(size_t)src[e] * (HID / 4) + q];
  float* o = acc + (size_t)d * HID + q * 4;
  atomicAdd(o + 0, hv.x * w);
  atomicAdd(o + 1, hv.y * w);
  atomicAdd(o + 2, hv.z * w);
  atomicAdd(o + 3, hv.w * w);
}

// ---------------- bias (+ optional ReLU), in place ----------------------------
__global__ void bias_act(float* __restrict__ acc, const float* __restrict__ b, int relu) {
  const int i = blockIdx.x * blockDim.x + threadIdx.x;
  if (i >= NNODES * HID) return;
  float v = acc[i] + b[i & (HID - 1)];
  if (relu) v = fmaxf(v, 0.f);
  acc[i] = v;
}

extern "C" void kernel_launch(void* const* d_in, const int* in_sizes, int n_in,
                              void* d_out, int out_size, void* d_ws, size_t ws_size,
                              hipStream_t stream) {
  // inputs (setup_inputs order)
  const float* x         = (const float*)d_in[0];
  const int*   src       = (const int*)d_in[1];
  const int*   dst       = src + NEDGES;
  const float* edge_attr = (const float*)d_in[2];
  const float* W1        = (const float*)d_in[3];
  const float* att_src1  = (const float*)d_in[4];
  const float* att_dst1  = (const float*)d_in[5];
  const float* We1       = (const float*)d_in[6];
  const float* att_e1    = (const float*)d_in[7];
  const float* b1        = (const float*)d_in[8];
  const float* W2        = (const float*)d_in[9];
  const float* att_src2  = (const float*)d_in[10];
  const float* att_dst2  = (const float*)d_in[11];
  const float* We2       = (const float*)d_in[12];
  const float* att_e2    = (const float*)d_in[13];
  const float* b2        = (const float*)d_in[14];
  float* out = (float*)d_out;

  // workspace layout
  float*    ws     = (float*)d_ws;
  float*    h      = ws;                        // 12.8M floats
  float*    acc1   = h + (size_t)NNODES * HID;  // 12.8M floats
  float*    a_src  = acc1 + (size_t)NNODES * HID;
  float*    a_dst  = a_src + NNODES;
  float*    alpha  = a_dst + NNODES;            // NEDGES
  float*    denom  = alpha + NEDGES;            // NNODES
  unsigned* amax   = (unsigned*)(denom + NNODES); // NNODES (contiguous after denom)
  float*    we_att = (float*)(amax + NNODES);   // 16 floats (layer1 @ 0, layer2 @ 8)

  const int waveBlocks   = (NNODES * 32) / 256;          // 6250 (exact)
  const int edgeBlocks   = NEDGES / 256;                 // 3125 (exact)
  const int scatBlocks   = (NEDGES * 64) / 256;          // 200000 (exact)
  const int accElems     = NNODES * HID;
  const int accBlocks    = (accElems + 255) / 256;

  edge_weight_proj<<<1, 32, 0, stream>>>(We1, att_e1, We2, att_e2, we_att);

  // ---------------- layer 1 ----------------
  gemm_wmma_f32<<<waveBlocks, 256, 0, stream>>>(x, W1, h, IN_DIM);
  node_dots<<<waveBlocks, 256, 0, stream>>>(h, att_src1, att_dst1, a_src, a_dst);
  zero_f32<<<(2 * NNODES + 255) / 256, 256, 0, stream>>>(denom, 2 * NNODES); // denom + amax
  zero_f32<<<accBlocks, 256, 0, stream>>>(acc1, accElems);
  edge_alpha<<<edgeBlocks, 256, 0, stream>>>(src, dst, edge_attr, we_att,
                                             a_src, a_dst, alpha, amax);
  edge_exp<<<edgeBlocks, 256, 0, stream>>>(dst, alpha, amax, denom);
  edge_scatter<<<scatBlocks, 256, 0, stream>>>(src, dst, alpha, denom, h, acc1);
  bias_act<<<accBlocks, 256, 0, stream>>>(acc1, b1, /*relu=*/1);

  // ---------------- layer 2 (input = acc1, output accumulates into d_out) ----
  gemm_wmma_f32<<<waveBlocks, 256, 0, stream>>>(acc1, W2, h, HID);
  node_dots<<<waveBlocks, 256, 0, stream>>>(h, att_src2, att_dst2, a_src, a_dst);
  zero_f32<<<(2 * NNODES + 255) / 256, 256, 0, stream>>>(denom, 2 * NNODES);
  zero_f32<<<accBlocks, 256, 0, stream>>>(out, accElems);
  edge_alpha<<<edgeBlocks, 256, 0, stream>>>(src, dst, edge_attr, we_att + 8,
                                             a_src, a_dst, alpha, amax);
  edge_exp<<<edgeBlocks, 256, 0, stream>>>(dst, alpha, amax, denom);
  edge_scatter<<<scatBlocks, 256, 0, stream>>>(src, dst, alpha, denom, h, out);
  bias_act<<<accBlocks, 256, 0, stream>>>(out, b2, /*relu=*/0);
}